// ImprovedCLIPLoss_22574348108228
// MI455X (gfx1250) — compile-verified
//
#include <hip/hip_runtime.h>
#include <math.h>
#include <stdint.h>

// ---------------------------------------------------------------------------
// Point-cloud rasterization (depth + normal maps), MI455X / gfx1250.
//  phase 0: init workspace (zero 5 accumulation planes, seed min/max slots)
//  phase 1: per-batch min/max of x,y,z  (ordered-uint atomics, deterministic)
//  phase 2: scatter-add w, zn*w, n*w into 5 planes (L2-resident f32 atomics),
//           point stream staged through LDS with async global->LDS copies
//  phase 3: per-pixel finalize, write [B,3,S,S], reduce global min/max
//  phase 4: in-place global min-max normalize of d_out
// ---------------------------------------------------------------------------

#define SDIM   224
#define EPSF   1e-8f
#define TPB    256
#define TILE   256
#define NBATCH 16

constexpr size_t kPix = (size_t)SDIM * SDIM;            // pixels per image
constexpr size_t kBP  = (size_t)NBATCH * kPix;          // elements per plane

// ---- CDNA5 async global->LDS path (guarded; fallback = direct loads) ------
#if defined(__HIP_DEVICE_COMPILE__) && defined(__gfx1250__) && \
    __has_builtin(__builtin_amdgcn_global_load_async_to_lds_b128)
#define HAS_ASYNC_PATH 1
#else
#define HAS_ASYNC_PATH 0
#endif

#if HAS_ASYNC_PATH
#if __has_builtin(__builtin_amdgcn_s_wait_asynccnt)
#define WAIT_ASYNC(n) __builtin_amdgcn_s_wait_asynccnt(n)
#else
#define WAIT_ASYNC(n) asm volatile("s_wait_asynccnt %0" ::"n"(n) : "memory")
#endif

typedef int v4i __attribute__((ext_vector_type(4)));
typedef __attribute__((address_space(1))) v4i* as1_v4i;   // ptr to global int4
typedef __attribute__((address_space(3))) v4i* as3_v4i;   // ptr to LDS int4

__device__ __forceinline__ void async_cp16(const float* g, float* l) {
  // 16B global -> LDS via the async copy engine (tracked by ASYNCcnt).
  __builtin_amdgcn_global_load_async_to_lds_b128(
      (as1_v4i)(v4i*)(void*)const_cast<float*>(g),
      (as3_v4i)(v4i*)(void*)l,
      0, 0);
}

// Stage one 256-point tile (points 768f | dens 256f | normals 768f = 1792f).
// Exactly TWO async instructions per wave (uniform), so WAIT_ASYNC(2) after
// issuing the *next* tile guarantees the *current* tile has landed.
__device__ __forceinline__ void stage_tile(const float* __restrict__ pt,
                                           const float* __restrict__ dt,
                                           const float* __restrict__ nt,
                                           float* __restrict__ lbuf, int tid) {
  // issue A: i in [0,256): 192 float4 of points, 64 float4 of densities
  {
    const float* g = (tid < 192) ? (pt + tid * 4) : (dt + (tid - 192) * 4);
    float*       l = (tid < 192) ? (lbuf + tid * 4) : (lbuf + 768 + (tid - 192) * 4);
    async_cp16(g, l);
  }
  // issue B: 192 float4 of normals (tid>=192 re-issues a duplicate, benign,
  // to keep ASYNCcnt increments wave-uniform at exactly 2 per stage)
  {
    int k = (tid < 192) ? tid : (tid - 64);
    async_cp16(nt + k * 4, lbuf + 1024 + k * 4);
  }
}
#endif  // HAS_ASYNC_PATH

// ---- ordered-uint encoding so float min/max can use integer atomics -------
__device__ __forceinline__ unsigned f2o(float f) {
  unsigned u = __float_as_uint(f);
  return (u & 0x80000000u) ? ~u : (u | 0x80000000u);
}
__device__ __forceinline__ float o2f(unsigned e) {
  unsigned u = (e & 0x80000000u) ? (e ^ 0x80000000u) : ~e;
  return __uint_as_float(u);
}
__device__ __forceinline__ int clampi(int v, int lo, int hi) {
  return v < lo ? lo : (v > hi ? hi : v);
}

// ---------------------------------------------------------------------------
__global__ void pc_init(unsigned* __restrict__ ws_u,
                        unsigned long long planeCount, int mmCount) {
  unsigned long long i = (unsigned long long)blockIdx.x * TPB + threadIdx.x;
  if (i < planeCount) {
    reinterpret_cast<float*>(ws_u)[i] = 0.0f;
  } else if (i < planeCount + (unsigned long long)mmCount) {
    int k = (int)(i - planeCount);                      // even = min, odd = max
    ws_u[i] = (k & 1) ? f2o(-__builtin_inff()) : f2o(__builtin_inff());
  }
}

// ---------------------------------------------------------------------------
__global__ void pc_minmax(const float* __restrict__ pts,
                          unsigned* __restrict__ mm, int N) {
  const int b = blockIdx.y;
  const int tid = threadIdx.x;
  const float* p = pts + (size_t)b * N * 3;
  const int stride = gridDim.x * TPB;

  float mn0 = __builtin_inff(), mn1 = mn0, mn2 = mn0;
  float mx0 = -__builtin_inff(), mx1 = mx0, mx2 = mx0;
  for (int i = blockIdx.x * TPB + tid; i < N; i += stride) {
    if (i + 4 * stride < N)
      __builtin_prefetch(p + (size_t)(i + 4 * stride) * 3, 0, 0);  // global_prefetch_b8
    float x = p[(size_t)i * 3 + 0];
    float y = p[(size_t)i * 3 + 1];
    float z = p[(size_t)i * 3 + 2];
    mn0 = fminf(mn0, x); mx0 = fmaxf(mx0, x);
    mn1 = fminf(mn1, y); mx1 = fmaxf(mx1, y);
    mn2 = fminf(mn2, z); mx2 = fmaxf(mx2, z);
  }

  __shared__ float smn[TPB], smx[TPB];
  float mns[3] = {mn0, mn1, mn2};
  float mxs[3] = {mx0, mx1, mx2};
  for (int c = 0; c < 3; ++c) {
    smn[tid] = mns[c];
    smx[tid] = mxs[c];
    __syncthreads();
    for (int o = TPB / 2; o > 0; o >>= 1) {
      if (tid < o) {
        smn[tid] = fminf(smn[tid], smn[tid + o]);
        smx[tid] = fmaxf(smx[tid], smx[tid + o]);
      }
      __syncthreads();
    }
    if (tid == 0) {
      atomicMin(&mm[b * 6 + 2 * c + 0], f2o(smn[0]));
      atomicMax(&mm[b * 6 + 2 * c + 1], f2o(smx[0]));
    }
    __syncthreads();
  }
}

// ---------------------------------------------------------------------------
__device__ __forceinline__ void scatter_point(
    float x, float y, float z, float d, float nx, float ny, float nz,
    float xmin, float xs, float ymin, float ys, float zmin, float zs,
    float* __restrict__ planes, size_t pixBase) {
  int xp = clampi((int)((x - xmin) * xs), 0, SDIM - 1);
  int yp = clampi((int)((y - ymin) * ys), 0, SDIM - 1);
  float zn = (z - zmin) * zs;
  float w = 1.0f / (1.0f + expf(-d));
  size_t pix = pixBase + (size_t)yp * SDIM + xp;
  atomicAdd(planes + pix, w);                 // wsum
  atomicAdd(planes + kBP + pix, zn * w);      // depth
  atomicAdd(planes + 2 * kBP + pix, nx * w);  // n.x
  atomicAdd(planes + 3 * kBP + pix, ny * w);  // n.y
  atomicAdd(planes + 4 * kBP + pix, nz * w);  // n.z
}

__global__ void pc_scatter(const float* __restrict__ pts,
                           const float* __restrict__ dens,
                           const float* __restrict__ nrms,
                           const unsigned* __restrict__ mm,
                           float* __restrict__ planes, int N, int fullTiles) {
  const int b = blockIdx.y;
  const int tid = threadIdx.x;
  float xmin = o2f(mm[b * 6 + 0]), xmax = o2f(mm[b * 6 + 1]);
  float ymin = o2f(mm[b * 6 + 2]), ymax = o2f(mm[b * 6 + 3]);
  float zmin = o2f(mm[b * 6 + 4]), zmax = o2f(mm[b * 6 + 5]);
  float xs = (float)(SDIM - 1) / (xmax - xmin + EPSF);
  float ys = (float)(SDIM - 1) / (ymax - ymin + EPSF);
  float zs = 1.0f / (zmax - zmin + EPSF);
  size_t pixBase = (size_t)b * kPix;
  const float* ptsB = pts + (size_t)b * N * 3;
  const float* densB = dens + (size_t)b * N;
  const float* nrmsB = nrms + (size_t)b * N * 3;

#if HAS_ASYNC_PATH
  __shared__ float buf[2][7 * TILE];  // 2 x 7168B, double-buffered tiles
  const int stride = gridDim.x;
  int t = blockIdx.x;
  int s = 0;
  if (t < fullTiles)
    stage_tile(ptsB + (size_t)t * TILE * 3, densB + (size_t)t * TILE,
               nrmsB + (size_t)t * TILE * 3, buf[0], tid);
  for (; t < fullTiles; t += stride) {
    int tn = t + stride;
    if (tn < fullTiles) {
      stage_tile(ptsB + (size_t)tn * TILE * 3, densB + (size_t)tn * TILE,
                 nrmsB + (size_t)tn * TILE * 3, buf[s ^ 1], tid);
      WAIT_ASYNC(2);   // next tile's 2 issues may still be in flight
    } else {
      WAIT_ASYNC(0);   // nothing new issued: drain current tile fully
    }
    __syncthreads();
    const float* B0 = buf[s];
    float x = B0[tid * 3 + 0], y = B0[tid * 3 + 1], z = B0[tid * 3 + 2];
    float d = B0[3 * TILE + tid];
    float nx = B0[4 * TILE + tid * 3 + 0];
    float ny = B0[4 * TILE + tid * 3 + 1];
    float nz = B0[4 * TILE + tid * 3 + 2];
    scatter_point(x, y, z, d, nx, ny, nz, xmin, xs, ymin, ys, zmin, zs,
                  planes, pixBase);
    __syncthreads();   // everyone done reading buf[s] before it is restaged
    s ^= 1;
  }
  // tail (partial tile): direct loads
  for (int i = fullTiles * TILE + blockIdx.x * TPB + tid; i < N;
       i += gridDim.x * TPB) {
    scatter_point(ptsB[(size_t)i * 3], ptsB[(size_t)i * 3 + 1],
                  ptsB[(size_t)i * 3 + 2], densB[i], nrmsB[(size_t)i * 3],
                  nrmsB[(size_t)i * 3 + 1], nrmsB[(size_t)i * 3 + 2], xmin, xs,
                  ymin, ys, zmin, zs, planes, pixBase);
  }
#else
  (void)fullTiles;
  for (int i = blockIdx.x * TPB + tid; i < N; i += gridDim.x * TPB) {
    scatter_point(ptsB[(size_t)i * 3], ptsB[(size_t)i * 3 + 1],
                  ptsB[(size_t)i * 3 + 2], densB[i], nrmsB[(size_t)i * 3],
                  nrmsB[(size_t)i * 3 + 1], nrmsB[(size_t)i * 3 + 2], xmin, xs,
                  ymin, ys, zmin, zs, planes, pixBase);
  }
#endif
}

// ---------------------------------------------------------------------------
__global__ void pc_finalize(const float* __restrict__ planes,
                            float* __restrict__ out,
                            unsigned* __restrict__ gmm, int total) {
  const int tid = threadIdx.x;
  int idx = blockIdx.x * TPB + tid;
  float lmin = __builtin_inff(), lmax = -__builtin_inff();
  if (idx < total) {
    int b = idx / (int)kPix;
    int r = idx - b * (int)kPix;
    float wsum = planes[idx];
    float dep = planes[kBP + idx];
    float nx = planes[2 * kBP + idx];
    float ny = planes[3 * kBP + idx];
    float nz = planes[4 * kBP + idx];
    bool valid = wsum > 0.0f;
    float inv = valid ? (1.0f / wsum) : 1.0f;
    dep = valid ? dep * inv : dep;
    nx = valid ? nx * inv : nx;
    ny = valid ? ny * inv : ny;
    nz = valid ? nz * inv : nz;
    float nrm = sqrtf(nx * nx + ny * ny + nz * nz + EPSF);
    float c1 = (nx / nrm + 1.0f) * 0.5f;
    float c2 = (ny / nrm + 1.0f) * 0.5f;
    size_t o = (size_t)b * 3 * kPix + r;        // [B, 3, S, S]
    out[o] = dep;
    out[o + kPix] = c1;
    out[o + 2 * kPix] = c2;
    lmin = fminf(dep, fminf(c1, c2));
    lmax = fmaxf(dep, fmaxf(c1, c2));
  }
  __shared__ float smn[TPB], smx[TPB];
  smn[tid] = lmin;
  smx[tid] = lmax;
  __syncthreads();
  for (int o = TPB / 2; o > 0; o >>= 1) {
    if (tid < o) {
      smn[tid] = fminf(smn[tid], smn[tid + o]);
      smx[tid] = fmaxf(smx[tid], smx[tid + o]);
    }
    __syncthreads();
  }
  if (tid == 0) {
    atomicMin(&gmm[0], f2o(smn[0]));
    atomicMax(&gmm[1], f2o(smx[0]));
  }
}

// ---------------------------------------------------------------------------
__global__ void pc_normalize(float* __restrict__ out,
                             const unsigned* __restrict__ gmm, int n) {
  int i = blockIdx.x * TPB + threadIdx.x;
  if (i < n) {
    float cmin = o2f(gmm[0]);
    float cmax = o2f(gmm[1]);
    out[i] = (out[i] - cmin) / (cmax - cmin + EPSF);
  }
}

// ---------------------------------------------------------------------------
extern "C" void kernel_launch(void* const* d_in, const int* in_sizes, int n_in,
                              void* d_out, int out_size, void* d_ws,
                              size_t ws_size, hipStream_t stream) {
  const float* pts = (const float*)d_in[0];   // [B, N, 3]
  const float* dens = (const float*)d_in[1];  // [B, N, 1]
  const float* nrms = (const float*)d_in[2];  // [B, N, 3]
  float* out = (float*)d_out;                 // [B, 3, S, S]
  (void)n_in;
  (void)ws_size;

  const int B = NBATCH;
  const int N = in_sizes[1] / B;

  // workspace: 5 planes (f32) | per-batch minmax (6*B u32) | global minmax (2 u32)
  unsigned* ws_u = (unsigned*)d_ws;
  float* planes = (float*)d_ws;
  const unsigned long long planeCount = 5ull * kBP;
  unsigned* mm = ws_u + planeCount;
  unsigned* gmm = mm + 6 * B;
  const int mmCount = 6 * B + 2;

  // phase 0: init
  {
    unsigned long long tot = planeCount + (unsigned long long)mmCount;
    unsigned blocks = (unsigned)((tot + TPB - 1) / TPB);
    pc_init<<<blocks, TPB, 0, stream>>>(ws_u, planeCount, mmCount);
  }
  // phase 1: per-batch min/max
  pc_minmax<<<dim3(100, B), TPB, 0, stream>>>(pts, mm, N);
  // phase 2: scatter (async-LDS tiled path needs 16B-aligned tile starts)
  int fullTiles = ((N & 3) == 0) ? (N / TILE) : 0;
  pc_scatter<<<dim3(112, B), TPB, 0, stream>>>(pts, dens, nrms, mm, planes, N,
                                               fullTiles);
  // phase 3: finalize + global min/max
  {
    int total = (int)(B * kPix);
    pc_finalize<<<(total + TPB - 1) / TPB, TPB, 0, stream>>>(planes, out, gmm,
                                                             total);
  }
  // phase 4: global normalize in place
  pc_normalize<<<(out_size + TPB - 1) / TPB, TPB, 0, stream>>>(out, gmm,
                                                               out_size);
}